// SynthonCompletionModel_38165079392556
// MI455X (gfx1250) — compile-verified
//
#include <hip/hip_runtime.h>
#include <hip/hip_bf16.h>

// ---------------------------------------------------------------------------
// RGCN (SynthonCompletion) forward for MI455X / gfx1250.
//
// Math restructuring:
//   agg[v] = sum_e (x[src_e] @ Wrel[type_e]) / deg[v]
//          = sum_r ( sum_{e: type=r} x[src_e]/deg[dst_e] ) @ Wrel[r]
// per relation: scatter normalized features into tmp[N,256], then dense
// [N,256]x[256,256] GEMM on the fp32 WMMA path (v_wmma_f32_16x16x4_f32)
// accumulating into acc[N,256]; then acc += x @ Wself; then relu(acc + b).
//
// Workspace (floats): x | x_next | tmp | acc | invdeg | gsum | gcnt
//   4*N*256 + N + G*768 + G  ~= 417 MB for N=100k, G=2000.
// ---------------------------------------------------------------------------

typedef __attribute__((ext_vector_type(2))) float v2f;
typedef __attribute__((ext_vector_type(8))) float v8f;

__global__ void rgcn_zero_kernel(float* __restrict__ p, size_t n4) {
  // n4 = count of float4s
  float4* p4 = (float4*)p;
  size_t i = (size_t)blockIdx.x * blockDim.x + threadIdx.x;
  size_t stride = (size_t)gridDim.x * blockDim.x;
  float4 z = make_float4(0.f, 0.f, 0.f, 0.f);
  for (; i < n4; i += stride) p4[i] = z;
}

__global__ void rgcn_zero_scalar_kernel(float* __restrict__ p, size_t n) {
  size_t i = (size_t)blockIdx.x * blockDim.x + threadIdx.x;
  size_t stride = (size_t)gridDim.x * blockDim.x;
  for (; i < n; i += stride) p[i] = 0.0f;
}

// deg accumulation: deg[dst] += 1
__global__ void rgcn_deg_kernel(const int* __restrict__ edge_dst, int E,
                                float* __restrict__ deg) {
  int e = blockIdx.x * blockDim.x + threadIdx.x;
  if (e < E) unsafeAtomicAdd(&deg[edge_dst[e]], 1.0f);
}

// in-place: deg -> 1/max(deg,1)
__global__ void rgcn_invdeg_kernel(float* __restrict__ deg, int N) {
  int n = blockIdx.x * blockDim.x + threadIdx.x;
  if (n < N) deg[n] = 1.0f / fmaxf(deg[n], 1.0f);
}

// x = node_feature + center_mask*flag_W[1] + flag_b ; also copy node_feature
// into node_context columns [L*D, L*D+D).  float4 over N*D.
__global__ void rgcn_flag_kernel(const float* __restrict__ nf,
                                 const float* __restrict__ fw1,
                                 const float* __restrict__ fb,
                                 const int* __restrict__ center,
                                 int N, int D, int ROW, int LD,
                                 float* __restrict__ x,
                                 float* __restrict__ out) {
  const int D4 = D >> 2;
  size_t total = (size_t)N * D4;
  size_t i = (size_t)blockIdx.x * blockDim.x + threadIdx.x;
  size_t stride = (size_t)gridDim.x * blockDim.x;
  for (; i < total; i += stride) {
    int n = (int)(i / D4);
    int d4 = (int)(i - (size_t)n * D4);
    float4 v = ((const float4*)nf)[i];
    *(float4*)&out[(size_t)n * ROW + LD + d4 * 4] = v;   // node_feature slice
    float4 b = ((const float4*)fb)[d4];
    float4 f;
    if (center[n]) {
      float4 w = ((const float4*)fw1)[d4];
      f = make_float4(v.x + b.x + w.x, v.y + b.y + w.y,
                      v.z + b.z + w.z, v.w + b.w + w.w);
    } else {
      f = make_float4(v.x + b.x, v.y + b.y, v.z + b.z, v.w + b.w);
    }
    ((float4*)x)[i] = f;
  }
}

// For edges of relation `rel`: tmp[dst] += x[src] * invdeg[dst].
// One wave per edge; float4 gathers, hardware fp32 atomic adds.
__global__ __launch_bounds__(256) void rgcn_edge_scatter_kernel(
    const float* __restrict__ x, const float* __restrict__ invdeg,
    const int* __restrict__ src, const int* __restrict__ dst,
    const int* __restrict__ typ, int rel, int E,
    float* __restrict__ tmp) {
  int e = blockIdx.x * 8 + (threadIdx.x >> 5);
  if (e >= E) return;
  if (typ[e] != rel) return;
  int lane = threadIdx.x & 31;
  int s = src[e];
  int d = dst[e];
  float scale = invdeg[d];
  const float4* xs = (const float4*)(x + (size_t)s * 256);
  float* td = tmp + (size_t)d * 256;
#pragma unroll
  for (int i = 0; i < 2; ++i) {
    int c4 = i * 32 + lane;               // float4 index within the row
    float4 v = xs[c4];
    float* t = td + c4 * 4;
    unsafeAtomicAdd(t + 0, v.x * scale);
    unsafeAtomicAdd(t + 1, v.y * scale);
    unsafeAtomicAdd(t + 2, v.z * scale);
    unsafeAtomicAdd(t + 3, v.w * scale);
  }
}

// C[N,256] (+)= A[N,256] @ W[256,256], fp32 WMMA.
// Block: 256 threads = 8 waves; stages W[:, colBase:colBase+64] in 64KB LDS,
// K-pair interleaved: lw[(k>>1)*128 + c*2 + (k&1)] = W[k][colBase+c], so a
// lane's B fragment {K=k0, k0+1} at one column is a single aligned ds_load_b64
// landing directly in an even VGPR pair (no register shuffles before WMMA).
// Each wave: 16 rows x 64 cols = 4 WMMA tiles; K-loop step 4.
__global__ __launch_bounds__(256) void rgcn_gemm_acc_kernel(
    const float* __restrict__ A, const float* __restrict__ W,
    float* __restrict__ C, int nRows, int beta) {
  __shared__ float lw[256 * 64];
  const int tid = threadIdx.x;
  const int colBase = blockIdx.y * 64;

  // Cooperative swizzled load of the 256x64 weight tile.
#pragma unroll
  for (int i = 0; i < 16; ++i) {
    int f = i * 1024 + tid * 4;
    int k = f >> 6;
    int c = f & 63;
    float4 w = *(const float4*)&W[(size_t)k * 256 + colBase + c];
    float* q = &lw[(k >> 1) * 128 + (k & 1) + c * 2];
    q[0] = w.x;
    q[2] = w.y;
    q[4] = w.z;
    q[6] = w.w;
  }
  __syncthreads();

  const int lane = tid & 31;
  const int wave = tid >> 5;
  const int m = lane & 15;      // M (for A) / N (for B,C,D) index within tile
  const int half = lane >> 4;   // lane-half selects K pair / row group
  const int row = blockIdx.x * 128 + wave * 16;
  if (row >= nRows) return;     // wave-uniform: EXEC stays all-ones for WMMA

  v8f acc0 = {}, acc1 = {}, acc2 = {}, acc3 = {};
  const float* arow = A + (size_t)(row + m) * 256;
  const float* bbase = &lw[half * 128 + m * 2];

#pragma unroll 4
  for (int kk = 0; kk < 256; kk += 4) {
    const int k0 = kk + 2 * half;
    // A fragment (16x4 f32): VGPR v holds K = 2*half + v  -> one b64 load
    v2f a = *(const v2f*)(arow + k0);
    // B fragments: one aligned ds_load_b64 each, +32 floats per column tile
    const float* bb = bbase + (kk >> 1) * 128;
    v2f b0 = *(const v2f*)(bb);
    v2f b1 = *(const v2f*)(bb + 32);
    v2f b2 = *(const v2f*)(bb + 64);
    v2f b3 = *(const v2f*)(bb + 96);
    acc0 = __builtin_amdgcn_wmma_f32_16x16x4_f32(false, a, false, b0, (short)0, acc0, false, false);
    acc1 = __builtin_amdgcn_wmma_f32_16x16x4_f32(false, a, false, b1, (short)0, acc1, false, false);
    acc2 = __builtin_amdgcn_wmma_f32_16x16x4_f32(false, a, false, b2, (short)0, acc2, false, false);
    acc3 = __builtin_amdgcn_wmma_f32_16x16x4_f32(false, a, false, b3, (short)0, acc3, false, false);
  }

  // C/D layout: VGPR j holds row M = j + 8*half, col N = m  (ISA 7.12.2)
  float* crow = C + (size_t)row * 256 + colBase;
  if (beta) {
#pragma unroll
    for (int j = 0; j < 8; ++j) {
      size_t o = (size_t)(j + 8 * half) * 256 + m;
      crow[o]      += acc0[j];
      crow[o + 16] += acc1[j];
      crow[o + 32] += acc2[j];
      crow[o + 48] += acc3[j];
    }
  } else {
#pragma unroll
    for (int j = 0; j < 8; ++j) {
      size_t o = (size_t)(j + 8 * half) * 256 + m;
      crow[o]      = acc0[j];
      crow[o + 16] = acc1[j];
      crow[o + 32] = acc2[j];
      crow[o + 48] = acc3[j];
    }
  }
}

// x_next = relu(acc + b); also write into node_enc slice of node_context.
__global__ void rgcn_finish_layer_kernel(const float* __restrict__ acc,
                                         const float* __restrict__ b,
                                         int N, int D, int ROW, int layer,
                                         float* __restrict__ xn,
                                         float* __restrict__ out) {
  const int D4 = D >> 2;
  size_t total = (size_t)N * D4;
  size_t i = (size_t)blockIdx.x * blockDim.x + threadIdx.x;
  size_t stride = (size_t)gridDim.x * blockDim.x;
  for (; i < total; i += stride) {
    int n = (int)(i / D4);
    int d4 = (int)(i - (size_t)n * D4);
    float4 a = ((const float4*)acc)[i];
    float4 bb = ((const float4*)b)[d4];
    float4 v = make_float4(fmaxf(a.x + bb.x, 0.f), fmaxf(a.y + bb.y, 0.f),
                           fmaxf(a.z + bb.z, 0.f), fmaxf(a.w + bb.w, 0.f));
    ((float4*)xn)[i] = v;
    *(float4*)&out[(size_t)n * ROW + (size_t)layer * D + d4 * 4] = v;
  }
}

// per-graph sums of node_enc (read back from d_out) + counts
__global__ __launch_bounds__(256) void rgcn_pool_accum_kernel(
    const float* __restrict__ out_nodes, const int* __restrict__ batch,
    float* __restrict__ gsum, float* __restrict__ gcnt, int LD, int ROW) {
  int n = blockIdx.x;
  int g = batch[n];
  const float* enc = out_nodes + (size_t)n * ROW;
  float* gs = gsum + (size_t)g * LD;
  for (int c4 = threadIdx.x; c4 < (LD >> 2); c4 += 256) {
    float4 v = ((const float4*)enc)[c4];
    float* t = gs + c4 * 4;
    unsafeAtomicAdd(t + 0, v.x);
    unsafeAtomicAdd(t + 1, v.y);
    unsafeAtomicAdd(t + 2, v.z);
    unsafeAtomicAdd(t + 3, v.w);
  }
  if (threadIdx.x == 0) unsafeAtomicAdd(&gcnt[g], 1.0f);
}

// graph_context[g] = [gsum/cnt | one_hot(reaction)]
__global__ __launch_bounds__(256) void rgcn_graph_finalize_kernel(
    const float* __restrict__ gsum, const float* __restrict__ gcnt,
    const int* __restrict__ reaction, int LD, int NRXN,
    float* __restrict__ gctx) {
  int g = blockIdx.x;
  float inv = 1.0f / fmaxf(gcnt[g], 1.0f);
  int CTX = LD + NRXN;
  for (int c = threadIdx.x; c < CTX; c += 256) {
    float v;
    if (c < LD) v = gsum[(size_t)g * LD + c] * inv;
    else        v = (reaction[g] == (c - LD)) ? 1.0f : 0.0f;
    gctx[(size_t)g * CTX + c] = v;
  }
}

// node_context[:, LD+D : LD+D+CTX) = graph_context[batch[n]]
__global__ __launch_bounds__(256) void rgcn_broadcast_kernel(
    const float* __restrict__ gctx, const int* __restrict__ batch,
    int LD, int D, int CTX, int ROW, float* __restrict__ out) {
  int n = blockIdx.x;
  const float* src = gctx + (size_t)batch[n] * CTX;
  float* dst = out + (size_t)n * ROW + LD + D;
  // CTX = 778 = 2 (mod 4): float2 copy
  for (int c2 = threadIdx.x; c2 < (CTX >> 1); c2 += 256) {
    float2 v = ((const float2*)src)[c2];
    dst[c2 * 2]     = v.x;
    dst[c2 * 2 + 1] = v.y;
  }
}

extern "C" void kernel_launch(void* const* d_in, const int* in_sizes, int n_in,
                              void* d_out, int out_size, void* d_ws, size_t ws_size,
                              hipStream_t stream) {
  const float* node_feature = (const float*)d_in[0];
  const float* flag_W       = (const float*)d_in[1];
  const float* flag_b       = (const float*)d_in[2];
  const float* W_self       = (const float*)d_in[3];
  const float* b_self       = (const float*)d_in[4];
  const float* W_rel        = (const float*)d_in[5];
  const int*   edge_src     = (const int*)d_in[6];
  const int*   edge_dst     = (const int*)d_in[7];
  const int*   edge_type    = (const int*)d_in[8];
  const int*   batch        = (const int*)d_in[9];
  const int*   reaction     = (const int*)d_in[10];
  const int*   center       = (const int*)d_in[11];

  const int D = in_sizes[2];                 // 256
  const int N = in_sizes[0] / D;             // 100000
  const int E = in_sizes[6];                 // 400000
  const int L = in_sizes[4] / D;             // 3
  const int R = in_sizes[5] / (L * D * D);   // 4
  const int G = in_sizes[10];                // 2000
  const int LD = L * D;                      // 768
  const long long rem = (long long)out_size - (long long)N * (2LL * LD + D)
                        - (long long)G * LD;
  const int NRXN = (int)(rem / (long long)(N + G));   // 10
  const int CTX = LD + NRXN;                 // 778
  const int ROW = LD + D + CTX;              // 1802

  float* out  = (float*)d_out;
  float* gctx = out + (size_t)N * ROW;       // graph_context region of d_out

  const size_t ND = (size_t)N * D;
  float* x      = (float*)d_ws;
  float* xn     = x + ND;
  float* tmp    = xn + ND;
  float* acc    = tmp + ND;
  float* invdeg = acc + ND;
  float* gsum   = invdeg + N;
  float* gcnt   = gsum + (size_t)G * LD;

  // degree normalization (same every layer)
  rgcn_zero_scalar_kernel<<<256, 256, 0, stream>>>(invdeg, (size_t)N);
  rgcn_zero_kernel<<<256, 256, 0, stream>>>(gsum, (size_t)G * LD / 4);
  rgcn_zero_scalar_kernel<<<32, 256, 0, stream>>>(gcnt, (size_t)G);
  rgcn_deg_kernel<<<(E + 255) / 256, 256, 0, stream>>>(edge_dst, E, invdeg);
  rgcn_invdeg_kernel<<<(N + 255) / 256, 256, 0, stream>>>(invdeg, N);

  // input projection + node_feature slice of the output
  rgcn_flag_kernel<<<2048, 256, 0, stream>>>(node_feature, flag_W + D, flag_b,
                                             center, N, D, ROW, LD, x, out);

  const dim3 ggrid((N + 127) / 128, D / 64);
  for (int l = 0; l < L; ++l) {
    for (int r = 0; r < R; ++r) {
      rgcn_zero_kernel<<<4096, 256, 0, stream>>>(tmp, ND / 4);
      rgcn_edge_scatter_kernel<<<(E + 7) / 8, 256, 0, stream>>>(
          x, invdeg, edge_src, edge_dst, edge_type, r, E, tmp);
      rgcn_gemm_acc_kernel<<<ggrid, 256, 0, stream>>>(
          tmp, W_rel + ((size_t)(l * R + r)) * D * D, acc, N, r > 0 ? 1 : 0);
    }
    rgcn_gemm_acc_kernel<<<ggrid, 256, 0, stream>>>(
        x, W_self + (size_t)l * D * D, acc, N, 1);
    rgcn_finish_layer_kernel<<<4096, 256, 0, stream>>>(
        acc, b_self + (size_t)l * D, N, D, ROW, l, xn, out);
    float* t = x; x = xn; xn = t;
  }

  // graph pooling, context, broadcast
  rgcn_pool_accum_kernel<<<N, 256, 0, stream>>>(out, batch, gsum, gcnt, LD, ROW);
  rgcn_graph_finalize_kernel<<<G, 256, 0, stream>>>(gsum, gcnt, reaction, LD,
                                                    NRXN, gctx);
  rgcn_broadcast_kernel<<<N, 256, 0, stream>>>(gctx, batch, LD, D, CTX, ROW, out);
}